// CBMSparseMatrixGCN_80247168959057
// MI455X (gfx1250) — compile-verified
//
#include <hip/hip_runtime.h>

typedef __attribute__((ext_vector_type(2))) float v2f;
typedef __attribute__((ext_vector_type(8))) float v8f;

#define HIDDEN 128

// ---------------------------------------------------------------------------
// Zero output accumulator and degree buffer (required every call: scatter
// accumulates with atomics and the harness does not re-clear between replays).
// ---------------------------------------------------------------------------
__global__ __launch_bounds__(256) void gcn_zero_kernel(float* __restrict__ out,
                                                       size_t n_out,
                                                       float* __restrict__ deg,
                                                       int n_deg) {
    size_t i = (size_t)blockIdx.x * blockDim.x + threadIdx.x;
    size_t stride = (size_t)gridDim.x * blockDim.x;
    for (size_t j = i; j < n_out; j += stride) out[j] = 0.0f;
    for (size_t j = i; j < (size_t)n_deg; j += stride) deg[j] = 0.0f;
}

// ---------------------------------------------------------------------------
// h = x @ W^T via V_WMMA_F32_16X16X4_F32 (exact fp32 matrix-core path).
// Block = 256 threads = 8 waves. Block b computes rows [16b, 16b+16);
// wave w computes the 16x16 tile at columns [16w, 16w+16). K-loop: 32 WMMAs.
//
// ISA lane layouts (32-bit, wave32):
//   A 16x4 : lanes 0-15 -> M=lane,  VGPR{0,1} = K{0,1}; lanes 16-31 -> K{2,3}
//   B 4x16 : lanes 0-15 -> N=lane,  VGPR{0,1} = K{0,1}; lanes 16-31 -> K{2,3}
//   C 16x16: lane<16 -> (M=v, N=lane); lane>=16 -> (M=v+8, N=lane-16)
// ---------------------------------------------------------------------------
__global__ __launch_bounds__(256) void gcn_gemm_wmma_kernel(
        const float* __restrict__ x, const float* __restrict__ W,
        float* __restrict__ h, int N) {
    const int lane    = threadIdx.x & 31;
    const int wave    = threadIdx.x >> 5;        // column tile 0..7
    const int halfSel = lane >> 4;               // 0: K0..1, 1: K2..3
    const int l       = lane & 15;

    const int rowBase = blockIdx.x * 16;
    const int nBase   = wave * 16;

    // Clamp row for loads so EXEC stays all-ones for WMMA on ragged tails.
    int rowLd = rowBase + l;
    if (rowLd >= N) rowLd = N - 1;

    const float* __restrict__ xrow = x + (size_t)rowLd * HIDDEN;
    const float* __restrict__ wrow = W + (size_t)(nBase + l) * HIDDEN; // B[k][n] = W[n][k]

    v8f c = {};
    #pragma unroll
    for (int k = 0; k < HIDDEN; k += 4) {
        const int ka = k + (halfSel << 1);       // even -> 8B-aligned float2
        v2f a = *(const v2f*)(xrow + ka);
        v2f b = *(const v2f*)(wrow + ka);
        c = __builtin_amdgcn_wmma_f32_16x16x4_f32(
                /*neg_a=*/false, a, /*neg_b=*/false, b,
                /*c_mod=*/(short)0, c, /*reuse_a=*/false, /*reuse_b=*/false);
    }

    float* __restrict__ hp = h + (size_t)rowBase * HIDDEN + nBase + l;
    #pragma unroll
    for (int v = 0; v < 8; ++v) {
        const int m = v + halfSel * 8;
        if (rowBase + m < N) hp[(size_t)m * HIDDEN] = c[v];
    }
}

// ---------------------------------------------------------------------------
// deg[col[e]] += 1  (float histogram; counts are small so fp32 is exact)
// ---------------------------------------------------------------------------
__global__ __launch_bounds__(256) void gcn_deg_kernel(const int* __restrict__ col,
                                                      float* __restrict__ deg,
                                                      int E) {
    int stride = gridDim.x * blockDim.x;
    for (int i = blockIdx.x * blockDim.x + threadIdx.x; i < E; i += stride) {
        atomicAdd(&deg[col[i]], 1.0f);
    }
}

// ---------------------------------------------------------------------------
// deg -> dis = (deg > 0) ? rsqrt(deg) : 0   (in place)
// ---------------------------------------------------------------------------
__global__ __launch_bounds__(256) void gcn_dis_kernel(float* __restrict__ deg, int N) {
    int i = blockIdx.x * blockDim.x + threadIdx.x;
    if (i < N) {
        float d = deg[i];
        deg[i] = (d > 0.0f) ? __frsqrt_rn(d) : 0.0f;
    }
}

// ---------------------------------------------------------------------------
// Scatter: one wave32 per edge; lane j moves float4 #j of the 128-wide row.
//   out[row[e], :] += dis[row[e]] * dis[col[e]] * h[col[e], :]
// 512 B coalesced gather + 512 B of f32 atomics per edge; resolves in L2
// (h and out both fit in the 192 MB L2).
// ---------------------------------------------------------------------------
__global__ __launch_bounds__(256) void gcn_scatter_kernel(
        const float* __restrict__ h, const float* __restrict__ dis,
        const int* __restrict__ row, const int* __restrict__ col,
        float* __restrict__ out, int E) {
    const int lane  = threadIdx.x & 31;
    const int wavesPerBlock = blockDim.x >> 5;
    const int wave  = blockIdx.x * wavesPerBlock + (threadIdx.x >> 5);
    const int nWaves = gridDim.x * wavesPerBlock;

    for (int e = wave; e < E; e += nWaves) {
        const int r = row[e];
        const int cI = col[e];
        const float norm = dis[r] * dis[cI];   // lane-uniform broadcast loads

        const float4 hv = ((const float4*)(h + (size_t)cI * HIDDEN))[lane];
        float* op = out + (size_t)r * HIDDEN + lane * 4;
        atomicAdd(op + 0, hv.x * norm);
        atomicAdd(op + 1, hv.y * norm);
        atomicAdd(op + 2, hv.z * norm);
        atomicAdd(op + 3, hv.w * norm);
    }
}

// ---------------------------------------------------------------------------
extern "C" void kernel_launch(void* const* d_in, const int* in_sizes, int n_in,
                              void* d_out, int out_size, void* d_ws, size_t ws_size,
                              hipStream_t stream) {
    const float* x  = (const float*)d_in[0];   // [N, 128]
    const float* W  = (const float*)d_in[1];   // [128, 128]
    const int*   ei = (const int*)d_in[2];     // [2, E]
    float* out = (float*)d_out;                // [N, 128]

    const int N = in_sizes[0] / HIDDEN;
    const int E = in_sizes[2] / 2;
    const int* rowIdx = ei;
    const int* colIdx = ei + E;

    // Workspace layout: h [N*128 f32] | deg/dis [N f32]
    float* h   = (float*)d_ws;
    float* deg = h + (size_t)N * HIDDEN;

    // 1) zero accumulators
    gcn_zero_kernel<<<2048, 256, 0, stream>>>(out, (size_t)N * HIDDEN, deg, N);

    // 2) h = x @ W^T (WMMA fp32)
    const int rowTiles = (N + 15) / 16;
    gcn_gemm_wmma_kernel<<<rowTiles, 256, 0, stream>>>(x, W, h, N);

    // 3) degree histogram
    gcn_deg_kernel<<<1024, 256, 0, stream>>>(colIdx, deg, E);

    // 4) dis = rsqrt(deg)
    gcn_dis_kernel<<<(N + 255) / 256, 256, 0, stream>>>(deg, N);

    // 5) normalized scatter-add aggregation
    gcn_scatter_kernel<<<4096, 256, 0, stream>>>(h, deg, rowIdx, colIdx, out, E);
}